// Leg_fit_12945031430350
// MI455X (gfx1250) — compile-verified
//
#include <hip/hip_runtime.h>

// Collapsed computation:
//   e[t]   = sum_{u>=t} d[u]*0.8^(u-t)          (reverse exp filter of dopamine)
//   X[k,i] = e[t]*z_bar[t,b,i],  Y[k,j]=psi[t,b,j],  k = t*B+b
//   leg_grads = X^T * Y   (K=1024, N=256)  via V_WMMA_F32_16X16X4_F32
//
// X/Y are stored K-pair-interleaved: pair index = t (since B==2, k=t*2+b),
// element (k,i) lives at pair[t][i].{x=b0, y=b1}.  This makes every WMMA
// A/B fragment a single contiguous global_load_b64 per lane.

typedef __attribute__((ext_vector_type(2))) float v2f;
typedef __attribute__((ext_vector_type(8))) float v8f;

#define TT 512
#define BB 2
#define NN 256
#define KK (TT * BB)

#define THRESHOLD 0.615f
#define DAMPENING 0.3f
#define DECAY     0.8f

// ---------------------------------------------------------------------------
// Kernel 1: dopamine d[t] (causal conv of class-neuron spikes with reward
// kernel) then reverse exponential filter -> e[t].  One block, 512 threads.
// ---------------------------------------------------------------------------
__global__ __launch_bounds__(TT)
void k_reward(const float* __restrict__ z, const int* __restrict__ cn_idx,
              float* __restrict__ e_out) {
  __shared__ float s_kern[TT];
  __shared__ float s_zc[TT];
  __shared__ float s_d[TT];

  const int t = threadIdx.x;
  const int cn = cn_idx[0];

  // reward kernel: kp - km
  float ts = (float)t * (1.0f / 500.0f);
  float kp = 1.379f * (ts / 0.2f) * __expf(1.0f - ts / 0.2f);
  float km = 0.27f * ts * __expf(1.0f - ts);
  s_kern[t] = kp - km;

  // zc[t] = sum_b z[b,t,cn]; zc[0] forced to 0 (reference excludes s=0 term)
  float zc = 0.0f;
  if (t > 0) {
    zc = z[(0 * TT + t) * NN + cn] + z[(1 * TT + t) * NN + cn];
  }
  s_zc[t] = zc;
  __syncthreads();

  // d[t] = sum_{s<=t} kern[t-s] * zc[s]   (kern[0]==0, zc[0]==0)
  float d = 0.0f;
  for (int s = 1; s <= t; ++s) d += s_kern[t - s] * s_zc[s];
  s_d[t] = d;
  __syncthreads();

  // e[t] = sum_{u>=t} d[u] * DECAY^(u-t)
  float acc = 0.0f, w = 1.0f;
  for (int u = t; u < TT; ++u) {
    acc += w * s_d[u];
    w *= DECAY;
    if (w < 1e-38f) break;  // geometric weight underflowed
  }
  e_out[t] = acc;
}

// ---------------------------------------------------------------------------
// Kernel 2: one thread per neuron n, running BOTH batch scans:
//   z_bar[t] = DECAY*z_bar[t-1] + z[t-1]
//   X2[t][n] = { e[t]*z_bar(b=0), e[t]*z_bar(b=1) }   (b64 coalesced store)
//   Y2[t][n] = { psi(b=0),        psi(b=1)        }
// 256 threads, 1 block.
// ---------------------------------------------------------------------------
__global__ __launch_bounds__(256)
void k_traces(const float* __restrict__ v, const float* __restrict__ z,
              const float* __restrict__ e, v2f* __restrict__ X2,
              v2f* __restrict__ Y2) {
  const int n = threadIdx.x;

  const float inv_thr = 1.0f / THRESHOLD;
  float zb0 = 0.0f, zb1 = 0.0f;
  float zp0 = 0.0f, zp1 = 0.0f;

  for (int t = 0; t < TT; ++t) {
    zb0 = DECAY * zb0 + zp0;                     // z_bar uses z[t-1]
    zb1 = DECAY * zb1 + zp1;
    const int s0 = (0 * TT + t) * NN + n;
    const int s1 = (1 * TT + t) * NN + n;
    zp0 = z[s0];
    zp1 = z[s1];
    float vs0 = (v[s0] - THRESHOLD) * inv_thr;
    float vs1 = (v[s1] - THRESHOLD) * inv_thr;
    float psi0 = DAMPENING * fmaxf(1.0f - fabsf(vs0), 0.0f) * inv_thr;
    float psi1 = DAMPENING * fmaxf(1.0f - fabsf(vs1), 0.0f) * inv_thr;
    const float et = e[t];
    v2f xo, yo;
    xo.x = et * zb0; xo.y = et * zb1;
    yo.x = psi0;     yo.y = psi1;
    X2[t * NN + n] = xo;                         // global_store_b64, coalesced
    Y2[t * NN + n] = yo;
  }
}

// ---------------------------------------------------------------------------
// Kernel 3: leg_grads = X^T * Y  (256x256, K=1024) via fp32 WMMA 16x16x4.
// One wave32 per 16x16 output tile; 256 tiles = 64 blocks x 4 waves.
// A-frag layout (16x4 f32): lane = M + 16*(K>>1), vgpr = K&1
//   -> lane (half,idx) needs (X[kk,i], X[kk+1,i]) = X2[kk>>1][i]  (one b64)
// B-frag layout (4x16 f32): lane = N + 16*(K>>1), vgpr = K&1  (mirrored)
// C/D layout  (16x16 f32): vgpr r -> M = r + 8*(lane>>4), N = lane&15.
// ---------------------------------------------------------------------------
__global__ __launch_bounds__(128)
void k_gemm(const v2f* __restrict__ X2, const v2f* __restrict__ Y2,
            float* __restrict__ out) {
  const int lane = threadIdx.x & 31;
  const int wave = threadIdx.x >> 5;
  const int tile = blockIdx.x * 4 + wave;        // 0..255
  const int i0 = (tile >> 4) << 4;               // output row (pre) base
  const int j0 = (tile & 15) << 4;               // output col (post) base
  const int half = lane >> 4;
  const int idx = lane & 15;

  const v2f* Xp = X2 + i0 + idx;                 // A pair for row i0+idx
  const v2f* Yp = Y2 + j0 + idx;                 // B pair for col j0+idx

  v8f acc = {};
#pragma unroll 8
  for (int k = 0; k < KK; k += 4) {
    const int tp = (k >> 1) + half;              // K-pair index for this half
    v2f a  = Xp[(size_t)tp * NN];                // single b64 load
    v2f bf = Yp[(size_t)tp * NN];                // single b64 load
    acc = __builtin_amdgcn_wmma_f32_16x16x4_f32(
        /*neg_a=*/false, a, /*neg_b=*/false, bf,
        /*c_mod=*/(short)0, acc, /*reuse_a=*/false, /*reuse_b=*/false);
  }

#pragma unroll
  for (int r = 0; r < 8; ++r) {
    const int m = r + 8 * half;
    out[(size_t)(i0 + m) * NN + (j0 + idx)] = acc[r];
  }
}

// ---------------------------------------------------------------------------
extern "C" void kernel_launch(void* const* d_in, const int* in_sizes, int n_in,
                              void* d_out, int out_size, void* d_ws, size_t ws_size,
                              hipStream_t stream) {
  const float* v  = (const float*)d_in[0];
  const float* z  = (const float*)d_in[1];
  const int*   cn = (const int*)d_in[2];
  float* out = (float*)d_out;

  // workspace layout: e[512] floats | X2[512*256] v2f | Y2[512*256] v2f (~2.1MB)
  float* e = (float*)d_ws;
  v2f* X2 = (v2f*)(e + TT);
  v2f* Y2 = X2 + (size_t)TT * NN;

  k_reward<<<1, TT, 0, stream>>>(z, cn, e);
  k_traces<<<1, 256, 0, stream>>>(v, z, e, X2, Y2);
  k_gemm<<<(NN / 16) * (NN / 16) / 4, 128, 0, stream>>>(X2, Y2, out);
}